// GAE_43353399886295
// MI455X (gfx1250) — compile-verified
//
#include <hip/hip_runtime.h>
#include <cstddef>

// ---------------------------------------------------------------------------
// GAE forward: 2x GCNConv (GEMM + edge scatter-add), VAE heads, KL, sample,
// z@z^T adjacency reconstruction, feature decode. All dense GEMMs use
// V_WMMA_F32_16X16X4_F32 (full fp32 precision, 16x16 tile per wave32).
// The z@z^T kernel is specialized: 16x80 strip per wave (A-fragment reuse)
// + non-temporal stores (400MB write-once output must not evict 192MB L2).
// ---------------------------------------------------------------------------

#define GN 10000   // nodes
#define GF 256     // input features
#define GO 128     // conv2 out
#define GL 64      // latent
#define GE 320000  // edges

typedef float v2f __attribute__((ext_vector_type(2)));
typedef float v8f __attribute__((ext_vector_type(8)));

__device__ __forceinline__ void atomic_add_f32(float* p, float v) {
  __hip_atomic_fetch_add(p, v, __ATOMIC_RELAXED, __HIP_MEMORY_SCOPE_AGENT);
}

// ---------------------------------------------------------------------------
// Generic fp32 WMMA GEMM:  C[M,N] = act(A[M,K] @ B + bias)
//   B is [K,N] row-major. One 16x16 output tile per wave.
//   M,N multiples of 16; K multiple of 4 (guaranteed by problem dims).
// ---------------------------------------------------------------------------
template <bool BIAS, bool RELU>
__global__ void wmma_gemm_f32(const float* __restrict__ A,
                              const float* __restrict__ B,
                              const float* __restrict__ bias,
                              float* __restrict__ C,
                              int M, int N, int K) {
  const int lane = threadIdx.x & 31;
  const int wave = threadIdx.x >> 5;
  const int tilesN = N >> 4;
  const int tilesM = M >> 4;
  int tile = blockIdx.x * (blockDim.x >> 5) + wave;
  if (tile >= tilesM * tilesN) return;   // wave-uniform: EXEC stays all-1s

  const int tm   = tile / tilesN;
  const int tn   = tile - tm * tilesN;
  const int half = lane >> 4;            // 0: lanes 0-15, 1: lanes 16-31
  const int l15  = lane & 15;
  const int m    = (tm << 4) + l15;      // A row this lane supplies
  const int n    = (tn << 4) + l15;      // C/B column this lane supplies

  const size_t arow = (size_t)m * K;
  v8f acc = {};

  for (int k0 = 0; k0 < K; k0 += 4) {
    const int bk = k0 + (half << 1);     // per ISA A/B 16x4 f32 VGPR layout
    v2f a = *(const v2f*)(A + arow + bk);          // 8B-aligned (arow,bk even)
    v2f b;
    b.x = B[(size_t)bk * N + n];                   // strided column gather
    b.y = B[(size_t)(bk + 1) * N + n];
    acc = __builtin_amdgcn_wmma_f32_16x16x4_f32(
        /*neg_a=*/false, a, /*neg_b=*/false, b,
        /*c_mod=*/(short)0, acc, /*reuse_a=*/false, /*reuse_b=*/false);
  }

  const float bv = BIAS ? bias[n] : 0.0f;
#pragma unroll
  for (int r = 0; r < 8; ++r) {          // D layout: VGPR r -> M = r + 8*half
    const int mr = (tm << 4) + r + (half << 3);
    float v = acc[r] + bv;
    if (RELU) v = v > 0.0f ? v : 0.0f;
    C[(size_t)mr * N + n] = v;
  }
}

// ---------------------------------------------------------------------------
// Specialized ADJ = Z @ Z^T, Z:[N,GL] row-major, output [N,N] (400 MB).
// Each wave computes a 16x80 strip (5 consecutive 16x16 tiles): the A
// fragment is loaded once per k-step and feeds 5 WMMAs. 625 N-tiles = 5*125.
// Stores are non-temporal: output is write-once streaming, > L2 capacity.
// ---------------------------------------------------------------------------
#define ZZT_NB 5
__global__ void __launch_bounds__(256) wmma_zzt(const float* __restrict__ Zm,
                                                float* __restrict__ ADJ) {
  const int lane = threadIdx.x & 31;
  const int wave = threadIdx.x >> 5;
  const int stripsN = (GN / 16) / ZZT_NB;            // 125
  const int nStrips = (GN / 16) * stripsN;           // 625*125 = 78125
  int strip = blockIdx.x * 8 + wave;
  if (strip >= nStrips) return;                      // wave-uniform

  const int tm   = strip / stripsN;
  const int tn0  = (strip - tm * stripsN) * ZZT_NB;
  const int half = lane >> 4;
  const int l15  = lane & 15;
  const int m    = (tm << 4) + l15;

  const size_t arow = (size_t)m * GL;
  size_t brow[ZZT_NB];
#pragma unroll
  for (int j = 0; j < ZZT_NB; ++j)
    brow[j] = (size_t)(((tn0 + j) << 4) + l15) * GL;

  v8f acc[ZZT_NB];
#pragma unroll
  for (int j = 0; j < ZZT_NB; ++j) acc[j] = (v8f){};

  for (int k0 = 0; k0 < GL; k0 += 4) {
    const int bk = k0 + (half << 1);
    const v2f a = *(const v2f*)(Zm + arow + bk);     // loaded once, used 5x
#pragma unroll
    for (int j = 0; j < ZZT_NB; ++j) {
      const v2f b = *(const v2f*)(Zm + brow[j] + bk);
      acc[j] = __builtin_amdgcn_wmma_f32_16x16x4_f32(
          false, a, false, b, (short)0, acc[j], false, false);
    }
  }

#pragma unroll
  for (int j = 0; j < ZZT_NB; ++j) {
    const int n = ((tn0 + j) << 4) + l15;
#pragma unroll
    for (int r = 0; r < 8; ++r) {
      const int mr = (tm << 4) + r + (half << 3);
      __builtin_nontemporal_store(acc[j][r], &ADJ[(size_t)mr * GN + n]);
    }
  }
}

// ---------------------------------------------------------------------------
// Edge message scatter-add: agg[dst[e], :] += H[src[e], :] * w[e]
// One edge per block, one feature per thread (d = 128 or 256).
// ---------------------------------------------------------------------------
__global__ void scatter_edges(const float* __restrict__ H,
                              const int* __restrict__ src,
                              const int* __restrict__ dst,
                              const float* __restrict__ ew,
                              float* __restrict__ agg, int d) {
  const int e = blockIdx.x;
  const int f = threadIdx.x;
  const int s = src[e];
  const int t = dst[e];
  const float w = ew[e];
  atomic_add_f32(&agg[(size_t)t * d + f], H[(size_t)s * d + f] * w);
}

// ---------------------------------------------------------------------------
// Elementwise: X[i] = act(X[i] + b[i & dmask])   (d is a power of two)
// ---------------------------------------------------------------------------
__global__ void bias_act(float* __restrict__ X, const float* __restrict__ b,
                         int dmask, size_t total, int relu) {
  size_t i = (size_t)blockIdx.x * blockDim.x + threadIdx.x;
  if (i >= total) return;
  float v = X[i] + b[i & (size_t)dmask];
  if (relu) v = fmaxf(v, 0.0f);
  X[i] = v;
}

__global__ void fill_zero(float* __restrict__ p, size_t n) {
  size_t i = (size_t)blockIdx.x * blockDim.x + threadIdx.x;
  if (i < n) p[i] = 0.0f;
}

// ---------------------------------------------------------------------------
// KL term accumulation + reparameterized sample.
//   kl += -0.5 * (1 + log(clip(var,1e-7,inf)) - mean^2 - var_pos)
//   z    = clip(mean,-10,10) + exp(0.5*clip(var,-10,10)) * eps
// ---------------------------------------------------------------------------
__global__ void kl_sample(const float* __restrict__ mean,
                          const float* __restrict__ var,
                          const float* __restrict__ eps,
                          float* __restrict__ z,
                          float* __restrict__ kl, size_t total) {
  size_t i = (size_t)blockIdx.x * blockDim.x + threadIdx.x;
  float term = 0.0f;
  if (i < total) {
    const float mu = mean[i];
    const float va = var[i];
    const float vp = fmaxf(va, 1e-7f);
    term = -0.5f * (1.0f + logf(vp) - mu * mu - vp);
    const float mc = fminf(fmaxf(mu, -10.0f), 10.0f);
    const float vc = fminf(fmaxf(va, -10.0f), 10.0f);
    z[i] = mc + expf(0.5f * vc) * eps[i];
  }
  // wave32 butterfly reduction, one atomic per wave
#pragma unroll
  for (int off = 16; off > 0; off >>= 1)
    term += __shfl_down(term, off, 32);
  if ((threadIdx.x & 31) == 0 && term != 0.0f) atomic_add_f32(kl, term);
}

// ---------------------------------------------------------------------------
// Host-side orchestration
// ---------------------------------------------------------------------------
static inline int cdiv_i(long a, long b) { return (int)((a + b - 1) / b); }

extern "C" void kernel_launch(void* const* d_in, const int* in_sizes, int n_in,
                              void* d_out, int out_size, void* d_ws, size_t ws_size,
                              hipStream_t stream) {
  const float* x   = (const float*)d_in[0];
  const int*   ei  = (const int*)d_in[1];     // [2, E]
  const float* ew  = (const float*)d_in[2];
  const float* W1  = (const float*)d_in[3];   // [F, 2O] = [256,256]
  const float* b1  = (const float*)d_in[4];
  const float* W2  = (const float*)d_in[5];   // [2O, O] = [256,128]
  const float* b2  = (const float*)d_in[6];
  const float* Wm  = (const float*)d_in[7];   // [O, L] = [128,64]
  const float* bm  = (const float*)d_in[8];
  const float* Wv  = (const float*)d_in[9];
  const float* bv  = (const float*)d_in[10];
  const float* Wd  = (const float*)d_in[11];  // [L, F] = [64,256]
  const float* bd  = (const float*)d_in[12];
  const float* eps = (const float*)d_in[13];
  (void)in_sizes; (void)n_in; (void)out_size; (void)ws_size;

  const int* src = ei;
  const int* dst = ei + GE;

  // workspace layout (fp32)
  float* ws   = (float*)d_ws;
  float* H1   = ws;                          // [N, 256]
  float* AGG1 = H1   + (size_t)GN * 256;     // [N, 256] -> h1 after bias+relu
  float* H2   = AGG1 + (size_t)GN * 256;     // [N, 128]
  float* AGG2 = H2   + (size_t)GN * 128;     // [N, 128] -> h after bias
  float* MEAN = AGG2 + (size_t)GN * 128;     // [N, 64]
  float* VAR  = MEAN + (size_t)GN * 64;      // [N, 64]

  // output layout (fp32): z | adj | feat | kl
  float* out      = (float*)d_out;
  float* Z        = out;                                   // N*L
  float* ADJ      = out + (size_t)GN * GL;                 // N*N
  float* FEAT     = ADJ + (size_t)GN * GN;                 // N*F
  float* KL       = FEAT + (size_t)GN * GF;                // 1

  const int TPB = 256;

  // zero the atomic-accumulated buffers
  fill_zero<<<cdiv_i((long)GN * 256, TPB), TPB, 0, stream>>>(AGG1, (size_t)GN * 256);
  fill_zero<<<cdiv_i((long)GN * 128, TPB), TPB, 0, stream>>>(AGG2, (size_t)GN * 128);
  fill_zero<<<1, 32, 0, stream>>>(KL, 1);

  // ---- conv1: H1 = x @ W1 ; scatter ; h1 = relu(agg + b1) ----
  {
    long tiles = (GN / 16) * (256 / 16);
    wmma_gemm_f32<false, false><<<cdiv_i(tiles, 8), TPB, 0, stream>>>(
        x, W1, nullptr, H1, GN, 256, GF);
  }
  scatter_edges<<<GE, 256, 0, stream>>>(H1, src, dst, ew, AGG1, 256);
  bias_act<<<cdiv_i((long)GN * 256, TPB), TPB, 0, stream>>>(
      AGG1, b1, 255, (size_t)GN * 256, 1);

  // ---- conv2: H2 = h1 @ W2 ; scatter ; h = agg + b2 ----
  {
    long tiles = (GN / 16) * (128 / 16);
    wmma_gemm_f32<false, false><<<cdiv_i(tiles, 8), TPB, 0, stream>>>(
        AGG1, W2, nullptr, H2, GN, 128, 256);
  }
  scatter_edges<<<GE, 128, 0, stream>>>(H2, src, dst, ew, AGG2, 128);
  bias_act<<<cdiv_i((long)GN * 128, TPB), TPB, 0, stream>>>(
      AGG2, b2, 127, (size_t)GN * 128, 0);

  // ---- heads: mean = h@Wm + bm ; var = h@Wv + bv ----
  {
    long tiles = (GN / 16) * (GL / 16);
    wmma_gemm_f32<true, false><<<cdiv_i(tiles, 8), TPB, 0, stream>>>(
        AGG2, Wm, bm, MEAN, GN, GL, GO);
    wmma_gemm_f32<true, false><<<cdiv_i(tiles, 8), TPB, 0, stream>>>(
        AGG2, Wv, bv, VAR, GN, GL, GO);
  }

  // ---- KL reduction + reparameterized sample z ----
  kl_sample<<<cdiv_i((long)GN * GL, TPB), TPB, 0, stream>>>(
      MEAN, VAR, eps, Z, KL, (size_t)GN * GL);

  // ---- adjacency reconstruction: ADJ = Z @ Z^T (store-bound, NT stores) ----
  {
    long strips = (long)(GN / 16) * ((GN / 16) / ZZT_NB);   // 78,125
    wmma_zzt<<<cdiv_i(strips, 8), TPB, 0, stream>>>(Z, ADJ);
  }

  // ---- feature decode: FEAT = relu(Z @ Wd + bd) ----
  {
    long tiles = (GN / 16) * (GF / 16);
    wmma_gemm_f32<true, true><<<cdiv_i(tiles, 8), TPB, 0, stream>>>(
        Z, Wd, bd, FEAT, GN, GF, GL);
  }
}